// RegionProposal_36129264894123
// MI455X (gfx1250) — compile-verified
//
#include <hip/hip_runtime.h>
#include <stdint.h>

// ---------------------------------------------------------------------------
// RPN proposal generation for MI455X (gfx1250, wave32).
// Pipeline: map -> radix-select top 6000 -> ordered compaction -> LDS bitonic
// sort -> gather -> WMMA-tiled IoU bitmask -> sequential greedy NMS -> output.
// ---------------------------------------------------------------------------

#define K_TOP     6000
#define KP        6016          // padded to multiple of 32 (and 16)
#define WORDS     188           // KP / 32 mask words per row
#define RTILES    376           // KP / 16 row tiles
#define IOU_THR_F 0.7f
#define MIN_SIZE_F 1.0e-3f
#define NEG_INF_SORT 0x007FFFFFu   // sortable(-inf)

typedef __attribute__((ext_vector_type(2))) float v2f;
typedef __attribute__((ext_vector_type(8))) float v8f;
typedef __attribute__((ext_vector_type(4))) int v4i;

#if __has_builtin(__builtin_amdgcn_wmma_f32_16x16x4_f32)
#define HAVE_WMMA4 1
#else
#define HAVE_WMMA4 0
#endif

#if __has_builtin(__builtin_amdgcn_global_load_async_to_lds_b128)
#define HAVE_ASYNC 1
#else
#define HAVE_ASYNC 0
#endif

#define AS1 __attribute__((address_space(1)))
#define AS3 __attribute__((address_space(3)))

#if HAVE_ASYNC
__device__ __forceinline__ void async_copy_b128(const void* g, void* l) {
  AS1 void* graw = (AS1 void*)(uintptr_t)g;   // global: flat addr == AS1 addr
  AS3 void* lraw = (AS3 void*)l;              // addrspacecast generic -> LDS
  __builtin_amdgcn_global_load_async_to_lds_b128((AS1 v4i*)graw, (AS3 v4i*)lraw, 0, 0);
}
__device__ __forceinline__ void wait_async_zero() {
#if __has_builtin(__builtin_amdgcn_s_wait_asynccnt)
  __builtin_amdgcn_s_wait_asynccnt(0);
#else
  asm volatile("s_wait_asynccnt 0x0" ::: "memory");
#endif
}
#endif

__device__ __forceinline__ unsigned sortable_f32(float f) {
  unsigned u = __float_as_uint(f);
  return (u & 0x80000000u) ? ~u : (u | 0x80000000u);
}

// ---------------------------------------------------------------------------
// Stage 1: decode boxes, validity, sortable keys. softmax is monotone in
// (cls1 - cls0) so we rank on the logit difference directly.
// ---------------------------------------------------------------------------
__global__ void k_map(const float* __restrict__ cls, const float* __restrict__ reg,
                      const float* __restrict__ anc, float4* __restrict__ boxes,
                      unsigned long long* __restrict__ keys, int n) {
  int i = blockIdx.x * blockDim.x + threadIdx.x;
  if (i >= n) return;
  float c0 = cls[2 * i], c1 = cls[2 * i + 1];
  float r0 = reg[4 * i], r1 = reg[4 * i + 1], r2 = reg[4 * i + 2], r3 = reg[4 * i + 3];
  float ax1 = anc[4 * i], ay1 = anc[4 * i + 1], ax2 = anc[4 * i + 2], ay2 = anc[4 * i + 3];
  float acx = (ax1 + ax2) * 0.5f, acy = (ay1 + ay2) * 0.5f;
  float aw = ax2 - ax1, ah = ay2 - ay1;
  float cx = r0 * aw + acx, cy = r1 * ah + acy;
  float w = expf(r2) * aw, h = expf(r3) * ah;
  float x1 = cx - w * 0.5f, y1 = cy - h * 0.5f;
  float x2 = cx + w * 0.5f, y2 = cy + h * 0.5f;
  x1 = fminf(fmaxf(x1, 0.0f), 1.0f);
  y1 = fminf(fmaxf(y1, 0.0f), 1.0f);
  x2 = fminf(fmaxf(x2, 0.0f), 1.0f);
  y2 = fminf(fmaxf(y2, 0.0f), 1.0f);
  float ws = x2 - x1, hs = y2 - y1;
  bool valid = (ws >= MIN_SIZE_F) && (hs >= MIN_SIZE_F);
  unsigned sk = valid ? sortable_f32(c1 - c0) : NEG_INF_SORT;
  boxes[i] = make_float4(x1, y1, x2, y2);
  // low 32 bits = ~index: for equal keys, larger composite == smaller index
  keys[i] = ((unsigned long long)sk << 32) | (unsigned)(~(unsigned)i);
}

// ---------------------------------------------------------------------------
// Radix select (4 x 8-bit digits, descending) to find the exact 6000th key.
// ---------------------------------------------------------------------------
__global__ void k_init(unsigned* __restrict__ bins, unsigned* __restrict__ state) {
  int t = blockIdx.x * blockDim.x + threadIdx.x;
  if (t < 256) bins[t] = 0u;
  if (t == 0) { state[0] = 0u; state[1] = (unsigned)K_TOP; }
}

__global__ void k_hist(const unsigned long long* __restrict__ keys, int n,
                       unsigned* __restrict__ bins, const unsigned* __restrict__ state,
                       int level) {
  __shared__ unsigned h[256];
  for (int t = threadIdx.x; t < 256; t += blockDim.x) h[t] = 0u;
  __syncthreads();
  unsigned prefix = state[0];
  int shift = 24 - 8 * level;
  for (int i = blockIdx.x * blockDim.x + threadIdx.x; i < n; i += gridDim.x * blockDim.x) {
    unsigned k32 = (unsigned)(keys[i] >> 32);
    if (level == 0 || (k32 >> (shift + 8)) == prefix)
      atomicAdd(&h[(k32 >> shift) & 255u], 1u);
  }
  __syncthreads();
  for (int t = threadIdx.x; t < 256; t += blockDim.x)
    if (h[t]) atomicAdd(&bins[t], h[t]);
}

__global__ void k_pick(unsigned* __restrict__ bins, unsigned* __restrict__ state) {
  if (threadIdx.x == 0 && blockIdx.x == 0) {
    unsigned kk = state[1];
    unsigned cum = 0u;
    int b = 255;
    for (; b > 0; --b) {
      if (cum + bins[b] >= kk) break;
      cum += bins[b];
    }
    state[0] = (state[0] << 8) | (unsigned)b;
    state[1] = kk - cum;
    for (int t = 0; t < 256; ++t) bins[t] = 0u;  // ready for next level
  }
}

// ---------------------------------------------------------------------------
// Ordered compaction (single workgroup, wave32 ballot scans). Takes key > T
// plus the E lowest-index keys == T -> exactly 6000 entries (set is exact and
// deterministic; order is fixed up by the sort that follows).
// ---------------------------------------------------------------------------
__device__ __forceinline__ unsigned block_rank(bool flag, unsigned* wTot, unsigned* total) {
  unsigned lane = threadIdx.x & 31u, wv = threadIdx.x >> 5;
  unsigned long long m = __ballot((int)flag);
  unsigned inWave = (unsigned)__popcll(m & ((1ull << lane) - 1ull));
  unsigned wt = (unsigned)__popcll(m);
  if (lane == 0) wTot[wv] = wt;
  __syncthreads();
  if (threadIdx.x == 0) {
    unsigned acc = 0u;
    for (int w = 0; w < 32; ++w) { unsigned t = wTot[w]; wTot[w] = acc; acc += t; }
    wTot[32] = acc;
  }
  __syncthreads();
  *total = wTot[32];
  unsigned r = wTot[wv] + inWave;
  __syncthreads();
  return r;  // exclusive rank among flagged threads
}

__global__ void __launch_bounds__(1024) k_compact(const unsigned long long* __restrict__ keys,
                                                  int n, const unsigned* __restrict__ state,
                                                  unsigned long long* __restrict__ sel) {
  __shared__ unsigned wTot[33];
  __shared__ unsigned sBase[2];  // running eq count, running taken count
  const unsigned T = state[0];
  const unsigned E = state[1];
  if (threadIdx.x == 0) { sBase[0] = 0u; sBase[1] = 0u; }
  __syncthreads();
  for (int base = 0; base < n; base += 1024) {
    unsigned eqBase = sBase[0];
    unsigned tBase = sBase[1];
    int i = base + (int)threadIdx.x;
    unsigned long long kv = 0ull;
    unsigned k32 = 0u;
    if (i < n) { kv = keys[i]; k32 = (unsigned)(kv >> 32); }
    bool gt = (i < n) && (k32 > T);
    bool eq = (i < n) && (k32 == T);
    unsigned eqTot, tTot;
    unsigned eqRank = block_rank(eq, wTot, &eqTot);
    bool take = gt || (eq && (eqBase + eqRank) < E);
    unsigned tRank = block_rank(take, wTot, &tTot);
    if (take) {
      unsigned pos = tBase + tRank;
      if (pos < (unsigned)K_TOP) sel[pos] = kv;
    }
    __syncthreads();
    if (threadIdx.x == 0) { sBase[0] = eqBase + eqTot; sBase[1] = tBase + tTot; }
    __syncthreads();
  }
}

// ---------------------------------------------------------------------------
// Bitonic sort of 6000 composite keys (pad to 8192) in 64 KB of LDS,
// descending: (score desc, index asc). One workgroup.
// ---------------------------------------------------------------------------
__global__ void __launch_bounds__(1024) k_sort(unsigned long long* __restrict__ sel) {
  __shared__ unsigned long long a[8192];
  for (int i = threadIdx.x; i < 8192; i += 1024) a[i] = (i < K_TOP) ? sel[i] : 0ull;
  __syncthreads();
  for (int k = 2; k <= 8192; k <<= 1) {
    for (int j = k >> 1; j > 0; j >>= 1) {
      for (int t = threadIdx.x; t < 4096; t += 1024) {
        int i = ((t & ~(j - 1)) << 1) | (t & (j - 1));
        int p = i | j;
        bool up = ((i & k) == 0);
        unsigned long long x = a[i], y = a[p];
        if ((up && x < y) || (!up && x > y)) { a[i] = y; a[p] = x; }
      }
      __syncthreads();
    }
  }
  for (int i = threadIdx.x; i < K_TOP; i += 1024) sel[i] = a[i];
}

// ---------------------------------------------------------------------------
// Gather the sorted top-6000 boxes + areas + validity (pad to 6016).
// ---------------------------------------------------------------------------
__global__ void k_gather(const unsigned long long* __restrict__ sel,
                         const float4* __restrict__ boxes, float4* __restrict__ b6k,
                         float* __restrict__ area, unsigned* __restrict__ validk, int n) {
  int i = blockIdx.x * blockDim.x + threadIdx.x;
  if (i >= KP) return;
  float4 b = make_float4(0.f, 0.f, 0.f, 0.f);
  unsigned v = 0u;
  if (i < K_TOP) {
    unsigned long long kv = sel[i];
    unsigned idx = ~((unsigned)kv);
    unsigned sk = (unsigned)(kv >> 32);
    if (idx < (unsigned)n) {
      b = boxes[idx];
      v = (sk != NEG_INF_SORT) ? 1u : 0u;
    }
  }
  b6k[i] = b;
  area[i] = (b.z - b.x) * (b.w - b.y);
  validk[i] = v;
}

// ---------------------------------------------------------------------------
// IoU bitmask kernel. One wave per (16-row x 32-col) mask block.
// WMMA f32 16x16x4 computes the pairwise area-sum tile as a rank-2 outer
// product: A = [area_i, 1, 0, 0], B = [1, area_j, 0, 0]^T.
// Column boxes are staged to LDS with async-to-LDS loads when available.
// ---------------------------------------------------------------------------
__global__ void __launch_bounds__(32) k_mask(const float4* __restrict__ b6k,
                                             const float* __restrict__ areaA,
                                             unsigned* __restrict__ mask) {
  const int c = blockIdx.x;   // column word (32 cols)
  const int rt = blockIdx.y;  // row tile (16 rows)
  const int rowBase = rt * 16;
  const int colBase = c * 32;
  if (colBase + 31 <= rowBase) return;  // no j > i bits possible in this block
  __shared__ float4 colB[32];
  __shared__ float4 rowB[16];
  __shared__ float aCol[32];
  __shared__ float aRow[16];
  __shared__ unsigned mw[16];
  const int lane = threadIdx.x;
  const int half = lane >> 4, l16 = lane & 15;
#if HAVE_ASYNC
  async_copy_b128(&b6k[colBase + lane], &colB[lane]);
#else
  colB[lane] = b6k[colBase + lane];
#endif
  aCol[lane] = areaA[colBase + lane];
  if (lane < 16) {
    rowB[lane] = b6k[rowBase + lane];
    aRow[lane] = areaA[rowBase + lane];
    mw[lane] = 0u;
  }
#if HAVE_ASYNC
  wait_async_zero();
#endif
  __syncthreads();

#pragma unroll
  for (int c16 = 0; c16 < 2; ++c16) {
    float sums[8];
#if HAVE_WMMA4
    // A 16x4 layout: lanes 0-15 hold row M=lane, K=0 (v0) / K=1 (v1);
    // lanes 16-31 hold K=2,3 -> zero. B 4x16: v+2*half selects K.
    v2f Aop, Bop;
    Aop.x = half ? 0.0f : aRow[l16];
    Aop.y = half ? 0.0f : 1.0f;
    Bop.x = half ? 0.0f : 1.0f;
    Bop.y = half ? 0.0f : aCol[c16 * 16 + l16];
    v8f Cz = {};
    v8f D = __builtin_amdgcn_wmma_f32_16x16x4_f32(false, Aop, false, Bop, (short)0, Cz,
                                                  false, false);
#pragma unroll
    for (int r = 0; r < 8; ++r) sums[r] = D[r];
#else
#pragma unroll
    for (int r = 0; r < 8; ++r) sums[r] = aRow[r + 8 * half] + aCol[c16 * 16 + l16];
#endif
#pragma unroll
    for (int r = 0; r < 8; ++r) {
      const int il = r + 8 * half;          // D-matrix M for (vgpr r, lane half)
      const int jl = c16 * 16 + l16;        // D-matrix N
      const float4 rb = rowB[il];
      const float4 cb = colB[jl];
      float iw = fminf(rb.z, cb.z) - fmaxf(rb.x, cb.x);
      float ih = fminf(rb.w, cb.w) - fmaxf(rb.y, cb.y);
      iw = fmaxf(iw, 0.0f);
      ih = fmaxf(ih, 0.0f);
      const float inter = iw * ih;
      const float iou = inter / (sums[r] - inter + 1e-9f);
      const int gi = rowBase + il, gj = colBase + jl;
      if (iou > IOU_THR_F && gj > gi) atomicOr(&mw[il], 1u << jl);
    }
  }
  __syncthreads();
  if (lane < 16) mask[(unsigned)(rowBase + lane) * WORDS + (unsigned)c] = mw[lane];
}

// ---------------------------------------------------------------------------
// Sequential greedy resolve (single workgroup) + final output.
// ---------------------------------------------------------------------------
__global__ void __launch_bounds__(256) k_nms(const unsigned* __restrict__ mask,
                                             const unsigned* __restrict__ validk,
                                             const float4* __restrict__ b6k,
                                             float* __restrict__ out) {
  __shared__ unsigned remv[WORDS];
  __shared__ unsigned keepw[WORDS];
  __shared__ unsigned rowword[32];
  __shared__ int selLds[300];
  __shared__ int keptCount;
  const int t = threadIdx.x;
  for (int w = t; w < WORDS; w += 256) { remv[w] = 0u; keepw[w] = 0u; }
  __syncthreads();
  for (int b = 0; b < WORDS; ++b) {
    if (t < 32) {
      int i = b * 32 + t;
      rowword[t] = (i < K_TOP) ? mask[(unsigned)i * WORDS + (unsigned)b] : 0u;
    }
    __syncthreads();
    if (t == 0) {
      unsigned cur = remv[b];
      unsigned kw = 0u;
      for (int j = 0; j < 32; ++j) {
        int i = b * 32 + j;
        if (i >= K_TOP) break;
        if (validk[i] && !((cur >> j) & 1u)) {
          kw |= (1u << j);
          cur |= rowword[j];  // in-word suppression (bits are strictly j' > i)
        }
      }
      keepw[b] = kw;
    }
    __syncthreads();
    unsigned kw = keepw[b];
    if (kw) {
      for (int w = b + 1 + t; w < WORDS; w += 256) {
        unsigned acc = remv[w];
        unsigned m = kw;
        while (m) {
          int j = __ffs(m) - 1;
          m &= m - 1u;
          acc |= mask[(unsigned)(b * 32 + j) * WORDS + (unsigned)w];
        }
        remv[w] = acc;
      }
    }
    __syncthreads();
  }
  if (t == 0) {
    int cnt = 0;
    for (int i = 0; i < K_TOP && cnt < 300; ++i)
      if ((keepw[i >> 5] >> (i & 31)) & 1u) selLds[cnt++] = i;
    keptCount = cnt;
  }
  __syncthreads();
  const int kc = keptCount;
  for (int r = t; r < 300; r += 256) {
    float4 b = (r < kc) ? b6k[selLds[r]] : make_float4(0.f, 0.f, 0.f, 0.f);
    out[4 * r + 0] = b.x;
    out[4 * r + 1] = b.y;
    out[4 * r + 2] = b.z;
    out[4 * r + 3] = b.w;
  }
}

// ---------------------------------------------------------------------------
extern "C" void kernel_launch(void* const* d_in, const int* in_sizes, int n_in,
                              void* d_out, int out_size, void* d_ws, size_t ws_size,
                              hipStream_t stream) {
  const float* cls = (const float*)d_in[0];
  const float* reg = (const float*)d_in[1];
  const float* anc = (const float*)d_in[2];
  float* out = (float*)d_out;
  const int n = in_sizes[0] / 2;  // 300000

  char* ws = (char*)d_ws;
  size_t off = 0;
  auto alloc = [&](size_t bytes) {
    void* p = ws + off;
    off = (off + bytes + 255) & ~(size_t)255;
    return p;
  };
  float4* boxes = (float4*)alloc((size_t)n * 16);
  unsigned long long* keys = (unsigned long long*)alloc((size_t)n * 8);
  unsigned long long* sel = (unsigned long long*)alloc((size_t)KP * 8);
  float4* b6k = (float4*)alloc((size_t)KP * 16);
  float* areaA = (float*)alloc((size_t)KP * 4);
  unsigned* validk = (unsigned*)alloc((size_t)KP * 4);
  unsigned* maskbuf = (unsigned*)alloc((size_t)KP * WORDS * 4);
  unsigned* bins = (unsigned*)alloc(256 * 4);
  unsigned* state = (unsigned*)alloc(4 * 4);

  k_map<<<(n + 255) / 256, 256, 0, stream>>>(cls, reg, anc, boxes, keys, n);
  k_init<<<1, 256, 0, stream>>>(bins, state);
  for (int lvl = 0; lvl < 4; ++lvl) {
    k_hist<<<512, 256, 0, stream>>>(keys, n, bins, state, lvl);
    k_pick<<<1, 1, 0, stream>>>(bins, state);
  }
  k_compact<<<1, 1024, 0, stream>>>(keys, n, state, sel);
  k_sort<<<1, 1024, 0, stream>>>(sel);
  k_gather<<<(KP + 255) / 256, 256, 0, stream>>>(sel, boxes, b6k, areaA, validk, n);
  k_mask<<<dim3(WORDS, RTILES), 32, 0, stream>>>(b6k, areaA, maskbuf);
  k_nms<<<1, 256, 0, stream>>>(maskbuf, validk, b6k, out);
}